// QCDDist_68813966017008
// MI455X (gfx1250) — compile-verified
//
#include <hip/hip_runtime.h>
#include <hip/hip_bf16.h>

// QCD pairwise-distance kernel for MI455X (gfx1250, wave32, WMMA).
//
// reference:  D[b,i,j] = (sq_i + sq_j - 2 * (E^T E)[i,j]) / R^2 * min(pm_i, pm_j)
//   E = emb[b]  [F=64, N=4096] (feature-major), sq_n = sum_f E[f][n]^2,
//   pm_n = exp(2*ALPHA*log(E[MOM][n]))
//
// Store-bandwidth bound (256 MB out vs 4 MB in): NT stores for output,
// fragment loads from global (L2-resident), V_WMMA_F32_16X16X4_F32 core.

typedef __attribute__((ext_vector_type(2))) float v2f;
typedef __attribute__((ext_vector_type(8))) float v8f;

namespace {
constexpr int   kF      = 64;
constexpr int   kMomCh  = 4;
constexpr float kAlpha  = 1.0f;
constexpr float kRadius = 0.5f;
}

// Block: 256 threads = 8 waves arranged 2 (row) x 4 (col).
// Each wave: 32x32 output (2x2 WMMA tiles). Block tile: 64 x 128.
__global__ __launch_bounds__(256) void qcd_dist_kernel(
    const float* __restrict__ emb, float* __restrict__ out, int N) {
  const int b = blockIdx.z;
  const float* __restrict__ Eb = emb + (size_t)b * kF * N;
  float* __restrict__ Ob = out + (size_t)b * N * (size_t)N;

  const int rowBase = blockIdx.y * 64;    // i-range of this block
  const int colBase = blockIdx.x * 128;   // j-range of this block

  // Epilogue vectors for this block's rows (64) and cols (128).
  __shared__ float s_sq[64 + 128];
  __shared__ float s_pm[64 + 128];

  const int tid = threadIdx.x;
  if (tid < 192) {
    const int n = (tid < 64) ? (rowBase + tid) : (colBase + (tid - 64));
    float sq = 0.0f;
#pragma unroll
    for (int f = 0; f < kF; ++f) {
      float v = Eb[(size_t)f * N + n];
      sq = fmaf(v, v, sq);
    }
    const float pt = Eb[(size_t)kMomCh * N + n];
    s_sq[tid] = sq;
    s_pm[tid] = __expf(2.0f * kAlpha * __logf(pt));
  }
  __syncthreads();

  const int wave = tid >> 5;        // 0..7
  const int lane = tid & 31;        // wave32
  const int wr   = wave >> 2;       // 0..1  wave row
  const int wc   = wave & 3;        // 0..3  wave col
  const int I0   = rowBase + wr * 32;
  const int J0   = colBase + wc * 32;
  const int r16  = lane & 15;       // position within 16-wide half
  const int kh   = lane >> 4;       // 0: K=0/1 (M low half of C), 1: K=2/3 (M high half)

  v8f acc[2][2];
#pragma unroll
  for (int ti = 0; ti < 2; ++ti)
#pragma unroll
    for (int tj = 0; tj < 2; ++tj)
      acc[ti][tj] = (v8f){0.f, 0.f, 0.f, 0.f, 0.f, 0.f, 0.f, 0.f};

  // K loop: 16 steps of K=4 over the 64 features.
#pragma unroll 4
  for (int kb = 0; kb < kF; kb += 4) {
    const int f0 = kb + 2 * kh;  // lanes 0-15: K=kb,kb+1 ; lanes 16-31: K=kb+2,kb+3
    v2f afrag[2], bfrag[2];
#pragma unroll
    for (int t = 0; t < 2; ++t) {
      const float* pa = Eb + (size_t)f0 * N + (I0 + 16 * t + r16);
      afrag[t].x = pa[0];
      afrag[t].y = pa[N];
      const float* pb = Eb + (size_t)f0 * N + (J0 + 16 * t + r16);
      bfrag[t].x = pb[0];
      bfrag[t].y = pb[N];
    }
#pragma unroll
    for (int ti = 0; ti < 2; ++ti)
#pragma unroll
      for (int tj = 0; tj < 2; ++tj)
        acc[ti][tj] = __builtin_amdgcn_wmma_f32_16x16x4_f32(
            /*neg_a=*/false, afrag[ti], /*neg_b=*/false, bfrag[tj],
            /*c_mod=*/(short)0, acc[ti][tj],
            /*reuse_a=*/false, /*reuse_b=*/false);
  }

  // Epilogue + streaming NT stores.
  const float invR2 = 1.0f / (kRadius * kRadius);
#pragma unroll
  for (int ti = 0; ti < 2; ++ti) {
#pragma unroll
    for (int tj = 0; tj < 2; ++tj) {
      const int lj  = wc * 32 + 16 * tj + r16;   // local col in [0,128)
      const float sqj = s_sq[64 + lj];
      const float pmj = s_pm[64 + lj];
      const size_t j  = (size_t)(colBase + lj);
#pragma unroll
      for (int r = 0; r < 8; ++r) {
        const int li  = wr * 32 + 16 * ti + r + 8 * kh;  // local row in [0,64)
        const float sqi = s_sq[li];
        const float pmi = s_pm[li];
        const float g   = acc[ti][tj][r];
        const float d   = (sqi + sqj - 2.0f * g) * invR2 * fminf(pmi, pmj);
        __builtin_nontemporal_store(d, &Ob[(size_t)(rowBase + li) * N + j]);
      }
    }
  }
}

extern "C" void kernel_launch(void* const* d_in, const int* in_sizes, int n_in,
                              void* d_out, int out_size, void* d_ws, size_t ws_size,
                              hipStream_t stream) {
  const float* emb = (const float*)d_in[0];
  float* out = (float*)d_out;

  const int B = 4;
  const int N = in_sizes[0] / (B * kF);  // 4096

  dim3 grid(N / 128, N / 64, B);
  qcd_dist_kernel<<<grid, 256, 0, stream>>>(emb, out, N);
}